// IAFF_Paper_43568148251205
// MI455X (gfx1250) — compile-verified
//
#include <hip/hip_runtime.h>
#include <math.h>

#define CCH   512
#define CINT  128
#define HWSZ  3136
#define NBAT  16
#define BNEPS 1e-5f

typedef float v2f __attribute__((ext_vector_type(2)));
typedef float v8f __attribute__((ext_vector_type(8)));

// ---------------- BN-fold: Weff = w * g/sqrt(v+eps); beff = b - m*g/sqrt(v+eps) ----
__global__ void fold_kernel(const float* __restrict__ w, const float* __restrict__ g,
                            const float* __restrict__ bt, const float* __restrict__ mu,
                            const float* __restrict__ var, float* __restrict__ Weff,
                            float* __restrict__ beff, int M, int K) {
    int tid = blockIdx.x * blockDim.x + threadIdx.x;
    if (tid >= M * K) return;
    int row = tid / K;
    float s = g[row] * rsqrtf(var[row] + BNEPS);
    Weff[tid] = w[tid] * s;
    if ((tid % K) == 0) beff[row] = bt[row] - mu[row] * s;
}

// ---------------- global-branch MLP on pooled vectors (tiny) ----------------------
__global__ void mlp_kernel(const float* __restrict__ pool, const float* __restrict__ W1,
                           const float* __restrict__ b1, const float* __restrict__ W2,
                           const float* __restrict__ b2, float* __restrict__ gatt) {
    __shared__ float xm[CCH];
    __shared__ float h[CINT];
    const int b = blockIdx.x;
    const int t = threadIdx.x;           // 128 threads
    for (int c = t; c < CCH; c += 128) xm[c] = pool[b * CCH + c] * (1.0f / (float)HWSZ);
    __syncthreads();
    {
        float acc = 0.f;
        const float* wr = W1 + (size_t)t * CCH;
        for (int c = 0; c < CCH; ++c) acc += wr[c] * xm[c];
        h[t] = fmaxf(acc + b1[t], 0.f);
    }
    __syncthreads();
    for (int o = t; o < CCH; o += 128) {
        float acc = 0.f;
        const float* wr = W2 + (size_t)o * CINT;
        for (int j = 0; j < CINT; ++j) acc += wr[j] * h[j];
        gatt[b * CCH + o] = acc + b2[o];
    }
}

// ---- CDNA5 async Global->LDS staging helpers (ASYNCcnt-tracked, no VGPR bounce) ----
__device__ __forceinline__ void async_copy_b128(unsigned lds_byte, const float* gsrc) {
    unsigned long long ga = (unsigned long long)(uintptr_t)gsrc;
    asm volatile("global_load_async_to_lds_b128 %0, %1, off"
                 :: "v"(lds_byte), "v"(ga) : "memory");
}
__device__ __forceinline__ void wait_async0() {
    asm volatile("s_wait_asynccnt 0" ::: "memory");
}

// ---------------- WMMA f32 GEMM: out[m,n] = sum_k A[m,k]*B[b,k,n] + bias[m] --------
// Block: 256 thr = 8 waves; tile M=128 x N=32; K in double-buffered LDS chunks of 32.
// A staged m-major via async b128 (stride 36: 16B-aligned writes, bank-spread b64
// frag reads). B staged n-major (stride 36) via float4 global load + LDS transpose so
// each B fragment is one contiguous ds_load_b64 (no register fix-up movs).
// MODE 0: write convy=out0, X=out1=convy+Y, pool-sum of X
// MODE 1: write H = relu(out)
// MODE 2: w=sigmoid(out+gatt); x' = Y + (convy-Y)*w -> out0; pool-sum of x'
// MODE 3: same as 2 but final store only (d_out)
#define KC  32
#define AST 36
#define BST 36

template <int MODE>
__global__ __launch_bounds__(256)
void gemm_kernel(const float* __restrict__ A, const float* __restrict__ Bact,
                 const float* __restrict__ bias, const float* __restrict__ Yin,
                 const float* __restrict__ convy_in, const float* __restrict__ gatt,
                 float* __restrict__ out0, float* __restrict__ out1,
                 float* __restrict__ pool, int M, int K) {
    __shared__ float As[2][128 * AST];   // m-major: As[m][k]
    __shared__ float Bs[2][32 * BST];    // n-major: Bs[n][k]
    const int tid    = threadIdx.x;
    const int lane   = tid & 31;
    const int wv     = tid >> 5;      // wave id -> M sub-tile
    const int laneHi = lane >> 4;
    const int l15    = lane & 15;
    const int b      = blockIdx.z;
    const int n0     = blockIdx.x * 32;
    const int mBlock = blockIdx.y * 128;

    v8f acc0 = {};
    v8f acc1 = {};

    // A chunk 128x32: 4 async b128 copies per thread (ASYNCcnt).
    auto issue_A = [&](int kc, int buf) {
        #pragma unroll
        for (int i = 0; i < 4; ++i) {
            int e  = tid + i * 256;        // 0..1023 quads
            int m  = e >> 3;               // 0..127
            int kq = (e & 7) << 2;         // 0..28 step 4
            async_copy_b128((unsigned)(uintptr_t)&As[buf][m * AST + kq],
                            A + (size_t)(mBlock + m) * K + kc + kq);
        }
    };
    // B chunk 32x32: one float4 (4 n for one k) per thread; transposed store later.
    const int bK  = tid >> 3;              // 0..31
    const int bNq = (tid & 7) << 2;        // 0..28 step 4
    float4 breg;
    auto load_B = [&](int kc) {
        breg = *(const float4*)(Bact + ((size_t)b * K + kc + bK) * HWSZ + n0 + bNq);
    };
    auto store_B = [&](int buf) {          // LDS transpose: Bs[n][k]
        Bs[buf][(bNq + 0) * BST + bK] = breg.x;
        Bs[buf][(bNq + 1) * BST + bK] = breg.y;
        Bs[buf][(bNq + 2) * BST + bK] = breg.z;
        Bs[buf][(bNq + 3) * BST + bK] = breg.w;
    };

    issue_A(0, 0);
    load_B(0);
    store_B(0);
    const int nch  = K / KC;
    const int mcol = wv * 16 + l15;        // A lane carries M
    for (int ic = 0; ic < nch; ++ic) {
        wait_async0();                     // this wave's A chunk landed
        __syncthreads();                   // all waves' A async + B ds_stores visible
        const int nxt = (ic + 1) & 1;
        if (ic + 1 < nch) {                // prefetch next chunk
            issue_A((ic + 1) * KC, nxt);
            load_B((ic + 1) * KC);
        }
        const int cur = ic & 1;
        const float* Ab = &As[cur][0];
        const float* Bb = &Bs[cur][0];
        #pragma unroll
        for (int kk = 0; kk < KC; kk += 4) {
            const int kA = kk + 2 * laneHi;
            v2f a   = *(const v2f*)&Ab[mcol * AST + kA];       // A: lane=M, v -> k
            v2f b0  = *(const v2f*)&Bb[l15 * BST + kA];        // B: lane=N, v -> k
            v2f b1v = *(const v2f*)&Bb[(16 + l15) * BST + kA];
            acc0 = __builtin_amdgcn_wmma_f32_16x16x4_f32(false, a, false, b0,
                                                         (short)0, acc0, false, false);
            acc1 = __builtin_amdgcn_wmma_f32_16x16x4_f32(false, a, false, b1v,
                                                         (short)0, acc1, false, false);
        }
        if (ic + 1 < nch) store_B(nxt);    // data has arrived during compute
    }

    // epilogue: C/D layout vgpr r -> m = r + 8*laneHi, n = lane&15 (+16 for acc1)
    const int mBase = mBlock + wv * 16 + 8 * laneHi;
    #pragma unroll
    for (int r = 0; r < 8; ++r) {
        const int m = mBase + r;
        const float bb = bias[m];
        const size_t i0 = ((size_t)b * M + m) * HWSZ + n0 + l15;
        float c0 = acc0[r] + bb;
        float c1 = acc1[r] + bb;
        if (MODE == 0) {
            out0[i0]      = c0;
            out0[i0 + 16] = c1;
            float x0 = c0 + Yin[i0];
            float x1 = c1 + Yin[i0 + 16];
            out1[i0]      = x0;
            out1[i0 + 16] = x1;
            float p = x0 + x1;
            p += __shfl_xor(p, 1);
            p += __shfl_xor(p, 2);
            p += __shfl_xor(p, 4);
            p += __shfl_xor(p, 8);
            if (l15 == 0) atomicAdd(&pool[b * CCH + m], p);
        } else if (MODE == 1) {
            out0[i0]      = fmaxf(c0, 0.f);
            out0[i0 + 16] = fmaxf(c1, 0.f);
        } else {  // MODE 2 / 3: sigmoid attention blend
            const float ga = gatt[b * CCH + m];
            float w0 = 1.f / (1.f + __expf(-(c0 + ga)));
            float w1 = 1.f / (1.f + __expf(-(c1 + ga)));
            float y0 = Yin[i0],      y1 = Yin[i0 + 16];
            float v0 = convy_in[i0], v1 = convy_in[i0 + 16];
            float o0 = y0 + (v0 - y0) * w0;
            float o1 = y1 + (v1 - y1) * w1;
            out0[i0]      = o0;
            out0[i0 + 16] = o1;
            if (MODE == 2) {
                float p = o0 + o1;
                p += __shfl_xor(p, 1);
                p += __shfl_xor(p, 2);
                p += __shfl_xor(p, 4);
                p += __shfl_xor(p, 8);
                if (l15 == 0) atomicAdd(&pool[b * CCH + m], p);
            }
        }
    }
}

extern "C" void kernel_launch(void* const* d_in, const int* in_sizes, int n_in,
                              void* d_out, int out_size, void* d_ws, size_t ws_size,
                              hipStream_t stream) {
    (void)in_sizes; (void)n_in; (void)out_size; (void)ws_size;
    const float* Y   = (const float*)d_in[0];
    const float* w_y = (const float*)d_in[1];
    const float* g_y = (const float*)d_in[2];
    const float* b_y = (const float*)d_in[3];
    const float* m_y = (const float*)d_in[4];
    const float* v_y = (const float*)d_in[5];
    // branches in dict order: ga(0), la(1), ga2(2), la2(3); each w1,g1,b1,m1,v1,w2,g2,b2,m2,v2
    const float* bp[4][10];
    for (int i = 0; i < 4; ++i)
        for (int j = 0; j < 10; ++j)
            bp[i][j] = (const float*)d_in[6 + i * 10 + j];

    // ---- workspace carve (floats) ----
    float* ws = (float*)d_ws;
    size_t off = 0;
    auto carve = [&](size_t n) { float* p = ws + off; off += n; return p; };
    float* Xbuf  = carve((size_t)NBAT * CCH * HWSZ);   // X, then reused as x'
    float* convY = carve((size_t)NBAT * CCH * HWSZ);
    float* Hbuf  = carve((size_t)NBAT * CINT * HWSZ);  // H1, then reused as H2
    float* WyE   = carve((size_t)CCH * CCH);
    float* byE   = carve(CCH);
    float *W1E[4], *b1E[4], *W2E[4], *b2E[4];
    for (int i = 0; i < 4; ++i) {
        W1E[i] = carve((size_t)CINT * CCH);
        b1E[i] = carve(CINT);
        W2E[i] = carve((size_t)CCH * CINT);
        b2E[i] = carve(CCH);
    }
    float* pool1 = carve(NBAT * CCH);
    float* pool2 = carve(NBAT * CCH);
    float* gatt1 = carve(NBAT * CCH);
    float* gatt2 = carve(NBAT * CCH);

    hipMemsetAsync(pool1, 0, 2 * NBAT * CCH * sizeof(float), stream);  // pool1+pool2

    // ---- fold BN into weights ----
    fold_kernel<<<(CCH * CCH + 255) / 256, 256, 0, stream>>>(w_y, g_y, b_y, m_y, v_y,
                                                             WyE, byE, CCH, CCH);
    for (int i = 0; i < 4; ++i) {
        fold_kernel<<<(CINT * CCH + 255) / 256, 256, 0, stream>>>(
            bp[i][0], bp[i][1], bp[i][2], bp[i][3], bp[i][4], W1E[i], b1E[i], CINT, CCH);
        fold_kernel<<<(CCH * CINT + 255) / 256, 256, 0, stream>>>(
            bp[i][5], bp[i][6], bp[i][7], bp[i][8], bp[i][9], W2E[i], b2E[i], CCH, CINT);
    }

    const dim3 blk(256);
    const dim3 gBig(HWSZ / 32, CCH / 128, NBAT);   // M=512
    const dim3 gSmall(HWSZ / 32, 1, NBAT);         // M=128

    // stage 0: conv_y, X = conv_y + Y, pooled sums of X
    gemm_kernel<0><<<gBig, blk, 0, stream>>>(WyE, Y, byE, Y, nullptr, nullptr,
                                             convY, Xbuf, pool1, CCH, CCH);
    // stage 1 attention
    mlp_kernel<<<NBAT, 128, 0, stream>>>(pool1, W1E[0], b1E[0], W2E[0], b2E[0], gatt1);
    gemm_kernel<1><<<gSmall, blk, 0, stream>>>(W1E[1], Xbuf, b1E[1], nullptr, nullptr,
                                               nullptr, Hbuf, nullptr, nullptr, CINT, CCH);
    gemm_kernel<2><<<gBig, blk, 0, stream>>>(W2E[1], Hbuf, b2E[1], Y, convY, gatt1,
                                             Xbuf, nullptr, pool2, CCH, CINT);  // x' -> Xbuf
    // stage 2 attention
    mlp_kernel<<<NBAT, 128, 0, stream>>>(pool2, W1E[2], b1E[2], W2E[2], b2E[2], gatt2);
    gemm_kernel<1><<<gSmall, blk, 0, stream>>>(W1E[3], Xbuf, b1E[3], nullptr, nullptr,
                                               nullptr, Hbuf, nullptr, nullptr, CINT, CCH);
    gemm_kernel<3><<<gBig, blk, 0, stream>>>(W2E[3], Hbuf, b2E[3], Y, convY, gatt2,
                                             (float*)d_out, nullptr, nullptr, CCH, CINT);
}